// AntiAliasInterpolation2d_72464688218420
// MI455X (gfx1250) — compile-verified
//
#include <hip/hip_runtime.h>
#include <stdint.h>

// ---------------------------------------------------------------------------
// Depthwise 13x13 Gaussian blur + nearest 4x downsample, fused & separable.
//   out[n,c,yo,xo] = sum_ky sum_kx w1[ky]*w1[kx] * x[n,c, 4yo-6+ky, 4xo-6+kx]
// Vertical pass on the matrix pipe (v_wmma_f32_16x16x4_f32, banded 16x76
// weight matrix), horizontal 13-tap pass on VALU. Input tiles staged to LDS
// with gfx1250 async global->LDS loads where available.
// ---------------------------------------------------------------------------

#define C_ 64
#define H_ 512
#define W_ 512
#define HO_ 128
#define WO_ 128

#define YT 16            // output rows per block  (WMMA M)
#define XT 32            // output cols per block
#define NROWS 76         // staged input rows: K padded to 19 wmma steps * 4
#define NV4 36           // float4 per staged row (144 cols = 9 N-tiles of 16)
#define NCOLS (NV4 * 4)
#define INSTRIDE 152     // floats; 2-row delta = 304 = 48 (mod 64) banks -> conflict-free half-wave split
#define TMPSTRIDE 148    // floats; 8-row delta = 1184 = 32 (mod 64) banks

typedef __attribute__((ext_vector_type(2))) float v2f;
typedef __attribute__((ext_vector_type(8))) float v8f;

// Async-load builtin takes pointers to GCC-vector int4 (per hipcc diagnostic):
//   param 0: '__attribute__((__vector_size__(4 * sizeof(int)))) int __device__ *'
typedef int v4i_g __attribute__((vector_size(16)));
typedef __attribute__((address_space(1))) v4i_g* gas_v4i_ptr;   // global (prints as __device__)
typedef __attribute__((address_space(3))) v4i_g* las_v4i_ptr;   // LDS

#if defined(__HIP_DEVICE_COMPILE__) && __has_builtin(__builtin_amdgcn_global_load_async_to_lds_b128)
#define USE_ASYNC_LDS 1
#else
#define USE_ASYNC_LDS 0
#endif

__global__ __launch_bounds__(256) void blur13_down4(
        const float* __restrict__ x, const float* __restrict__ w2d,
        float* __restrict__ out)
{
    __shared__ __align__(16) float sIn[NROWS * INSTRIDE];   // 46208 B
    __shared__ __align__(16) float sTmp[YT * TMPSTRIDE];    //  9472 B
    __shared__ float sW[16];

    const int tid  = threadIdx.x;
    const int lane = tid & 31;          // wave32
    const int wv   = tid >> 5;          // 8 waves
    const int nc   = blockIdx.x;        // 0..511  (n*C + c)
    const int yo0  = blockIdx.y * YT;   // output row base
    const int xo0  = blockIdx.z * XT;   // output col base
    const int Ybase  = 4 * yo0 - 6;     // global input row of staged row 0
    const int XbaseA = 4 * xo0 - 8;     // global input col of staged col 0 (16B aligned)

    // ---- 1D gaussian from the provided (separable) 2D kernel: row 6 / rowsum ----
    float wreg[13];
    {
        float s = 0.f;
        #pragma unroll
        for (int j = 0; j < 13; ++j) { wreg[j] = w2d[6 * 13 + j]; s += wreg[j]; }
        const float inv = 1.f / s;
        #pragma unroll
        for (int j = 0; j < 13; ++j) wreg[j] *= inv;
    }
    if (tid < 13) sW[tid] = wreg[tid];

    // ---- stage zero-padded input tile into LDS (async where available) ----
    const float* gbase = x + (size_t)nc * (H_ * W_);
    for (int idx = tid; idx < NROWS * NV4; idx += 256) {
        const int row = idx / NV4;
        const int q   = idx - row * NV4;
        const int y   = Ybase + row;
        const int xg  = XbaseA + 4 * q;       // multiple of 4 -> vec4 fully in or out
        float* lp = &sIn[row * INSTRIDE + 4 * q];
        if (((unsigned)y < (unsigned)H_) && ((unsigned)xg < (unsigned)W_)) {
            const float* gp = gbase + (size_t)y * W_ + xg;
#if USE_ASYNC_LDS
            __builtin_amdgcn_global_load_async_to_lds_b128(
                (gas_v4i_ptr)(uintptr_t)(const void*)gp,
                (las_v4i_ptr)(uint32_t)(uintptr_t)(void*)lp,
                0, 0);
#else
            *(float4*)lp = *(const float4*)gp;
#endif
        } else {
            *(float4*)lp = make_float4(0.f, 0.f, 0.f, 0.f);
        }
    }
#if USE_ASYNC_LDS
#if __has_builtin(__builtin_amdgcn_s_wait_asynccnt)
    __builtin_amdgcn_s_wait_asynccnt(0);
#else
    asm volatile("s_wait_asynccnt 0" ::: "memory");
#endif
#endif
    __syncthreads();

    // ---- vertical pass: Tmp[16 x 16] = Wv[16 x 76] * In[76 x 16] via WMMA ----
    // A layout (32-bit 16x4): lane L holds M=L&15; VGPR0/1 = K=kb, kb+1 with
    // kb = (L<16) ? 0 : 2.  B assumed symmetric: lane L holds N=L&15, rows kb, kb+1.
    const int mA  = lane & 15;
    const int kbA = (lane >> 4) * 2;
    v2f afrag[19];
    #pragma unroll
    for (int s = 0; s < 19; ++s) {
        const int j0 = 4 * s + kbA - 4 * mA;        // tap index w1[r - 4m]
        afrag[s].x = ((unsigned)j0 < 13u) ? sW[j0] : 0.f;
        afrag[s].y = ((unsigned)(j0 + 1) < 13u) ? sW[j0 + 1] : 0.f;
    }
    for (int t = wv; t < NCOLS / 16; t += 8) {      // 9 N-tiles over 8 waves
        const int x0 = t * 16;
        v8f acc = {0.f, 0.f, 0.f, 0.f, 0.f, 0.f, 0.f, 0.f};
        #pragma unroll
        for (int s = 0; s < 19; ++s) {
            const int rb = 4 * s + kbA;
            v2f b;
            b.x = sIn[(rb    ) * INSTRIDE + x0 + mA];
            b.y = sIn[(rb + 1) * INSTRIDE + x0 + mA];
            acc = __builtin_amdgcn_wmma_f32_16x16x4_f32(
                      false, afrag[s], false, b, (short)0, acc, false, false);
        }
        const int rbase = (lane >> 4) * 8;          // D: lanes<16 -> M=i, else M=i+8
        #pragma unroll
        for (int i = 0; i < 8; ++i)
            sTmp[(rbase + i) * TMPSTRIDE + x0 + mA] = acc[i];
    }
    __syncthreads();

    // ---- horizontal 13-tap pass (VALU) + coalesced store ----
    const int dxo = tid & 31;
    int yo = tid >> 5;                               // 0..7, then +8
    float* obase = out + ((size_t)nc * HO_ + yo0) * WO_ + xo0;
    #pragma unroll
    for (int rep = 0; rep < 2; ++rep, yo += 8) {
        const float* tp = &sTmp[yo * TMPSTRIDE + 4 * dxo + 2];  // x = 4*xo - 6 + k
        float acc = 0.f;
        #pragma unroll
        for (int k = 0; k < 13; ++k) acc = fmaf(wreg[k], tp[k], acc);
        obase[(size_t)yo * WO_ + dxo] = acc;
    }
}

extern "C" void kernel_launch(void* const* d_in, const int* in_sizes, int n_in,
                              void* d_out, int out_size, void* d_ws, size_t ws_size,
                              hipStream_t stream) {
    const float* x   = (const float*)d_in[0];   // (8, 64, 512, 512) fp32
    const float* w2d = (const float*)d_in[1];   // (64, 1, 13, 13)   fp32
    float* out = (float*)d_out;                 // (8, 64, 128, 128) fp32
    dim3 grid(8 * C_, HO_ / YT, WO_ / XT);      // (512, 8, 4)
    blur13_down4<<<grid, 256, 0, stream>>>(x, w2d, out);
}